// LinearNormalized_24275155157264
// MI455X (gfx1250) — compile-verified
//
#include <hip/hip_runtime.h>

// Quaternion (group-of-4) L2 normalization, fp32 -> fp32.
// Memory-bound streaming kernel: 512 MB total traffic, ~22 us floor at 23.3 TB/s.
// Strategy:
//  - 128-bit per-lane NT loads/stores (wave32 x 16B = 512B per VMEM instr)
//  - lane-private reduction (no LDS, no cross-lane)
//  - grid-stride with 4x unroll -> 4 outstanding global_load_b128 per lane
//  - hardware v_rsq_f32 (__builtin_amdgcn_rsqf) instead of IEEE divide chain
//    (op is bandwidth-bound; trims ~12 VALU/quaternion of issue-slot overhead)

typedef float v4f __attribute__((ext_vector_type(4)));

__device__ __forceinline__ float inv_norm(float n2) {
    // norm==0  <=>  n2==0 ; reference replaces zero norm with 1.0
    // guard also avoids rsq(0)=inf -> 0*inf=NaN
    return (n2 > 0.0f) ? __builtin_amdgcn_rsqf(n2) : 1.0f;
}

__global__ __launch_bounds__(256) void
quat_normalize_kernel(const v4f* __restrict__ x, v4f* __restrict__ out, int nquat) {
    int tid    = blockIdx.x * blockDim.x + threadIdx.x;
    int stride = gridDim.x * blockDim.x;

    int i = tid;
    for (; i + 3 * stride < nquat; i += 4 * stride) {
        v4f q0 = __builtin_nontemporal_load(&x[i]);
        v4f q1 = __builtin_nontemporal_load(&x[i + stride]);
        v4f q2 = __builtin_nontemporal_load(&x[i + 2 * stride]);
        v4f q3 = __builtin_nontemporal_load(&x[i + 3 * stride]);

        float n0 = q0.x * q0.x + q0.y * q0.y + q0.z * q0.z + q0.w * q0.w;
        float n1 = q1.x * q1.x + q1.y * q1.y + q1.z * q1.z + q1.w * q1.w;
        float n2 = q2.x * q2.x + q2.y * q2.y + q2.z * q2.z + q2.w * q2.w;
        float n3 = q3.x * q3.x + q3.y * q3.y + q3.z * q3.z + q3.w * q3.w;

        float s0 = inv_norm(n0);
        float s1 = inv_norm(n1);
        float s2 = inv_norm(n2);
        float s3 = inv_norm(n3);

        __builtin_nontemporal_store(q0 * s0, &out[i]);
        __builtin_nontemporal_store(q1 * s1, &out[i + stride]);
        __builtin_nontemporal_store(q2 * s2, &out[i + 2 * stride]);
        __builtin_nontemporal_store(q3 * s3, &out[i + 3 * stride]);
    }
    // Tail
    for (; i < nquat; i += stride) {
        v4f q = __builtin_nontemporal_load(&x[i]);
        float n = q.x * q.x + q.y * q.y + q.z * q.z + q.w * q.w;
        float s = inv_norm(n);
        __builtin_nontemporal_store(q * s, &out[i]);
    }
}

extern "C" void kernel_launch(void* const* d_in, const int* in_sizes, int n_in,
                              void* d_out, int out_size, void* d_ws, size_t ws_size,
                              hipStream_t stream) {
    (void)n_in; (void)d_ws; (void)ws_size; (void)out_size;

    const v4f* x  = (const v4f*)d_in[0];   // fp32, 65536 x 1024 flat
    v4f* out      = (v4f*)d_out;           // fp32, same shape
    int nelem     = in_sizes[0];           // 67,108,864 floats
    int nquat     = nelem >> 2;            // 16,777,216 groups of 4

    // 256 threads/block (8 wave32). Grid sized so each thread runs the 4x
    // unrolled body exactly once per loop trip: 4 quats (64B in / 64B out)
    // per thread per iteration.
    const int threads = 256;
    int blocks = (nquat + threads * 4 - 1) / (threads * 4);
    if (blocks < 1) blocks = 1;

    quat_normalize_kernel<<<blocks, threads, 0, stream>>>(x, out, nquat);
}